// TransformerEncoderLayer_83519934038296
// MI455X (gfx1250) — compile-verified
//
#include <hip/hip_runtime.h>
#include <hip/hip_bf16.h>

// ---------------------------------------------------------------------------
// Transformer encoder layer for MI455X (gfx1250), fp32, wave32.
//
// Roofline: pos_enc stream (256MB, read once) dominates -> ~11us at 23.3TB/s;
// total FLOPs only ~3.8G -> any WMMA rate suffices, so we use the *exact*
// fp32 V_WMMA_F32_16X16X4_F32 for every GEMM (no precision sacrifice needed).
//
// f32 WMMA per-lane layouts (ISA 7.12.2):
//   A (16x4):  row = lane%16,  a[j] = A[row][2*(lane>>4) + j]
//   B (4x16):  col = lane%16,  b[j] = B[2*(lane>>4) + j][col]
//   C/D (16x16): c[r] = C[r + 8*(lane>>4)][lane%16]
//
// The bandwidth kernel (posbias) uses GLOBAL_LOAD_ASYNC_TO_LDS_B128 with
// double buffering (ASYNCcnt-pipelined) when the toolchain exposes it.
// ---------------------------------------------------------------------------

typedef float v2f __attribute__((ext_vector_type(2)));
typedef float v8f __attribute__((ext_vector_type(8)));

#define DEPTH 64
#define NHEAD 8
#define SEQ   512
#define BATCH 4
#define BS    (BATCH * SEQ)   // 2048
#define FFDIM 128

#if defined(__has_builtin)
#if __has_builtin(__builtin_amdgcn_global_load_async_to_lds_b128) && \
    __has_builtin(__builtin_amdgcn_s_wait_asynccnt)
#define USE_ASYNC_LDS 1
#endif
#endif

__device__ __forceinline__ v8f wmma4(v2f a, v2f b, v8f c) {
  return __builtin_amdgcn_wmma_f32_16x16x4_f32(false, a, false, b, (short)0, c,
                                               false, false);
}
__device__ __forceinline__ v8f vzero8() {
  v8f z;
#pragma unroll
  for (int i = 0; i < 8; ++i) z[i] = 0.f;
  return z;
}
__device__ __forceinline__ v2f mkv2(float x, float y) {
  v2f r; r.x = x; r.y = y; return r;
}
__device__ __forceinline__ float hsum16(float v) {
#pragma unroll
  for (int m = 1; m < 16; m <<= 1) v += __shfl_xor(v, m, 32);
  return v;
}
__device__ __forceinline__ float hmax16(float v) {
#pragma unroll
  for (int m = 1; m < 16; m <<= 1) v = fmaxf(v, __shfl_xor(v, m, 32));
  return v;
}

// ---------------------------------------------------------------------------
// Kernel 1: Q/K/V projections.  [2048,64] @ [64,512] -> stored [B,h,S,d].
// ---------------------------------------------------------------------------
__global__ __launch_bounds__(32) void qkv_kernel(
    const float* __restrict__ seq, const float* __restrict__ Wq,
    const float* __restrict__ Wk, const float* __restrict__ Wv,
    float* __restrict__ q, float* __restrict__ k, float* __restrict__ v) {
  const int lane = threadIdx.x & 31;
  const int col = lane & 15, row = lane & 15, hi = lane >> 4;
  const int m0 = blockIdx.x * 16, n0 = blockIdx.y * 16;
  const int which = blockIdx.z;
  const float* W = (which == 0) ? Wq : (which == 1) ? Wk : Wv;
  float* out = (which == 0) ? q : (which == 1) ? k : v;
  const float scale = (which == 0) ? 0.125f : 1.0f;  // fold 1/sqrt(64) into q

  v8f c = vzero8();
#pragma unroll
  for (int kk = 0; kk < 16; ++kk) {
    const int kb = 4 * kk + 2 * hi;
    v2f a = *(const v2f*)(seq + (m0 + row) * DEPTH + kb);
    v2f b = mkv2(W[(kb + 0) * (DEPTH * NHEAD) + n0 + col],
                 W[(kb + 1) * (DEPTH * NHEAD) + n0 + col]);
    c = wmma4(a, b, c);
  }
  const int C = n0 + col;
  const int hh = C >> 6, dd = C & 63;
#pragma unroll
  for (int r = 0; r < 8; ++r) {
    const int R = m0 + r + 8 * hi;
    const int bb = R >> 9, s = R & 511;
    out[((bb * NHEAD + hh) * SEQ + s) * DEPTH + dd] = c[r] * scale;
  }
}

// ---------------------------------------------------------------------------
// Kernel 2: positional bias p[b,h,i,j] = pos_enc[b,i,j,:] @ Wp + bp.
// GEMM view: [B*S*S, 64] @ [64, 8 (zero-padded to 16)].
// Streams the 256MB pos_enc exactly once (the runtime floor).
// ---------------------------------------------------------------------------

// Common epilogue: D-tile (rows=j, cols=head) -> p[b,h,i,j] contiguous in j.
__device__ __forceinline__ void posbias_store(int t0, const v8f& c,
                                              const float* __restrict__ bp,
                                              float* __restrict__ p, int col,
                                              int hi) {
  const int bb = t0 >> 18;        // / (512*512)
  const int i = (t0 >> 9) & 511;  // / 512 % 512
  const int j0 = t0 & 511;
  if (col < NHEAD) {
    const float bias = bp[col];
    float* dst =
        p + (((size_t)(bb * NHEAD + col) * SEQ + i) * SEQ) + j0 + 8 * hi;
#pragma unroll
    for (int r = 0; r < 8; ++r) dst[r] = c[r] + bias;
  }
}

#if defined(USE_ASYNC_LDS)
// ---- Async path: GLOBAL_LOAD_ASYNC_TO_LDS_B128 double-buffered pipeline ----
// Builtin signature (from the round-2 diagnostic): param0 is
// `int __vector(4) addrspace(1)*`, so match it exactly.
typedef int vs4i __attribute__((vector_size(16)));
typedef __attribute__((address_space(1))) vs4i g_vs4i;
typedef __attribute__((address_space(3))) vs4i l_vs4i;

#define POS_TILES_PER_BLOCK 8
#define PB_LDS_STRIDE 68  // floats; 272B rows -> conflict-free b64 reads

__device__ __forceinline__ void posbias_prefetch(const float* __restrict__ pos,
                                                 int tile, float* lbuf,
                                                 int lane) {
  // One 16x64 fp32 tile = 4KB contiguous in memory; 8 coalesced 512B moves.
  const char* gbase = (const char*)pos + (size_t)tile * 4096;
#pragma unroll
  for (int cch = 0; cch < 8; ++cch) {
    const int off = cch * 512 + lane * 16;               // byte offset in tile
    const int ldsoff = (off >> 8) * 272 + (off & 255);   // pad 16B per 256B row
    __builtin_amdgcn_global_load_async_to_lds_b128(
        (g_vs4i*)(gbase + off), (l_vs4i*)((char*)lbuf + ldsoff), 0, 0);
  }
}

#define WAIT_ASYNC(n)                         \
  do {                                        \
    __builtin_amdgcn_s_wait_asynccnt(n);      \
    asm volatile("" ::: "memory");            \
  } while (0)

__global__ __launch_bounds__(32) void posbias_kernel(
    const float* __restrict__ pos, const float* __restrict__ Wp,
    const float* __restrict__ bp, float* __restrict__ p) {
  __shared__ float ab[2][16 * PB_LDS_STRIDE];
  const int lane = threadIdx.x & 31;
  const int col = lane & 15, row = lane & 15, hi = lane >> 4;
  const int tile_base = blockIdx.x * POS_TILES_PER_BLOCK;

  // Preload B (Wp) into registers once per block, zero-padding heads 8..15.
  v2f wp[16];
#pragma unroll
  for (int kk = 0; kk < 16; ++kk) {
    const int kb = 4 * kk + 2 * hi;
    wp[kk] = (col < NHEAD)
                 ? mkv2(Wp[(kb + 0) * NHEAD + col], Wp[(kb + 1) * NHEAD + col])
                 : mkv2(0.f, 0.f);
  }

  posbias_prefetch(pos, tile_base, &ab[0][0], lane);

  for (int t = 0; t < POS_TILES_PER_BLOCK; ++t) {
    // Prefetch next tile into the other buffer (its last reads finished a
    // full iteration ago -> no WAR hazard), then wait for current tile only
    // (async loads complete in order, so asynccnt<=8 means tile t landed).
    if (t + 1 < POS_TILES_PER_BLOCK) {
      posbias_prefetch(pos, tile_base + t + 1, &ab[(t + 1) & 1][0], lane);
      WAIT_ASYNC(8);
    } else {
      WAIT_ASYNC(0);
    }
    const float* a0 = &ab[t & 1][0];
    v8f c = vzero8();
#pragma unroll
    for (int kk = 0; kk < 16; ++kk) {
      const int kb = 4 * kk + 2 * hi;
      v2f a = *(const v2f*)(a0 + row * PB_LDS_STRIDE + kb);
      c = wmma4(a, wp[kk], c);
    }
    posbias_store((tile_base + t) * 16, c, bp, p, col, hi);
  }
}
#else
// ---- Fallback path: direct non-temporal strided loads --
__global__ __launch_bounds__(32) void posbias_kernel(
    const float* __restrict__ pos, const float* __restrict__ Wp,
    const float* __restrict__ bp, float* __restrict__ p) {
  const int lane = threadIdx.x & 31;
  const int col = lane & 15, row = lane & 15, hi = lane >> 4;
  const int t0 = blockIdx.x * 16;

  v2f wp[16];
#pragma unroll
  for (int kk = 0; kk < 16; ++kk) {
    const int kb = 4 * kk + 2 * hi;
    wp[kk] = (col < NHEAD)
                 ? mkv2(Wp[(kb + 0) * NHEAD + col], Wp[(kb + 1) * NHEAD + col])
                 : mkv2(0.f, 0.f);
  }

  v8f c = vzero8();
#pragma unroll
  for (int kk = 0; kk < 16; ++kk) {
    const int kb = 4 * kk + 2 * hi;
    v2f a = __builtin_nontemporal_load(
        (const v2f*)(pos + (size_t)(t0 + row) * DEPTH + kb));
    c = wmma4(a, wp[kk], c);
  }
  posbias_store(t0, c, bp, p, col, hi);
}
#endif

// ---------------------------------------------------------------------------
// Kernel 3: flash attention per (b,h) and 16-row query tile.
// ---------------------------------------------------------------------------
__global__ __launch_bounds__(32) void attn_kernel(
    const float* __restrict__ q, const float* __restrict__ k,
    const float* __restrict__ v, const float* __restrict__ p,
    float* __restrict__ o) {
  __shared__ float elds[16 * 17];
  const int lane = threadIdx.x & 31;
  const int col = lane & 15, row = lane & 15, hi = lane >> 4;
  const int i0 = blockIdx.x * 16;
  const int bh = blockIdx.y;

  const float* qb = q + (size_t)bh * SEQ * DEPTH;
  const float* kb_ = k + (size_t)bh * SEQ * DEPTH;
  const float* vb = v + (size_t)bh * SEQ * DEPTH;
  const float* pb = p + (size_t)bh * SEQ * SEQ;

  v2f aq[16];
#pragma unroll
  for (int kk = 0; kk < 16; ++kk)
    aq[kk] = *(const v2f*)(qb + (i0 + row) * DEPTH + 4 * kk + 2 * hi);

  float mrow[8], lrow[8];
#pragma unroll
  for (int r = 0; r < 8; ++r) { mrow[r] = -1e30f; lrow[r] = 0.f; }
  v8f acc[4] = {vzero8(), vzero8(), vzero8(), vzero8()};

  for (int j0 = 0; j0 < SEQ; j0 += 16) {
    v8f s = vzero8();
#pragma unroll
    for (int kk = 0; kk < 16; ++kk) {
      const int kb = 4 * kk + 2 * hi;
      v2f b = *(const v2f*)(kb_ + (j0 + col) * DEPTH + kb);
      s = wmma4(aq[kk], b, s);
    }
#pragma unroll
    for (int r = 0; r < 8; ++r)
      s[r] += pb[(size_t)(i0 + r + 8 * hi) * SEQ + j0 + col];

    float sc[8];
#pragma unroll
    for (int r = 0; r < 8; ++r) {
      const float mx = hmax16(s[r]);
      const float mn = fmaxf(mrow[r], mx);
      sc[r] = __expf(mrow[r] - mn);
      mrow[r] = mn;
    }
#pragma unroll
    for (int r = 0; r < 8; ++r) {
      const float e = __expf(s[r] - mrow[r]);
      lrow[r] = lrow[r] * sc[r] + hsum16(e);
      s[r] = e;
    }
#pragma unroll
    for (int t = 0; t < 4; ++t)
#pragma unroll
      for (int r = 0; r < 8; ++r) acc[t][r] *= sc[r];

    __syncthreads();
#pragma unroll
    for (int r = 0; r < 8; ++r) elds[(r + 8 * hi) * 17 + col] = s[r];
    __syncthreads();

#pragma unroll
    for (int kk = 0; kk < 4; ++kk) {
      const int kb = 4 * kk + 2 * hi;
      v2f a = mkv2(elds[row * 17 + kb], elds[row * 17 + kb + 1]);
#pragma unroll
      for (int t = 0; t < 4; ++t) {
        v2f b = mkv2(vb[(j0 + kb + 0) * DEPTH + t * 16 + col],
                     vb[(j0 + kb + 1) * DEPTH + t * 16 + col]);
        acc[t] = wmma4(a, b, acc[t]);
      }
    }
  }

  const int bb = bh >> 3, hh = bh & 7;
#pragma unroll
  for (int t = 0; t < 4; ++t)
#pragma unroll
    for (int r = 0; r < 8; ++r) {
      const int R = i0 + r + 8 * hi;
      o[((size_t)bb * SEQ + R) * (DEPTH * NHEAD) + hh * DEPTH + t * 16 + col] =
          acc[t][r] / lrow[r];
    }
}

// ---------------------------------------------------------------------------
// Kernel 4: fused epilogue per 16-row block:
//   x = LN(seq + heads @ Wo); y = relu(x@W1 + b1)@W2 + b2; out = LN(x + y).
// ---------------------------------------------------------------------------
__global__ __launch_bounds__(32) void epilogue_kernel(
    const float* __restrict__ o, const float* __restrict__ seq,
    const float* __restrict__ Wo, const float* __restrict__ W1,
    const float* __restrict__ b1, const float* __restrict__ W2,
    const float* __restrict__ b2, const float* __restrict__ g_att,
    const float* __restrict__ b_att, const float* __restrict__ g_ff,
    const float* __restrict__ b_ff, float* __restrict__ out) {
  __shared__ float xlds[16 * 68];
  __shared__ float ylds[16 * 132];
  const int lane = threadIdx.x & 31;
  const int col = lane & 15, row = lane & 15, hi = lane >> 4;
  const int m0 = blockIdx.x * 16;

  v8f xs[4] = {vzero8(), vzero8(), vzero8(), vzero8()};
  for (int kk = 0; kk < 128; ++kk) {
    const int kb = 4 * kk + 2 * hi;
    v2f a = *(const v2f*)(o + (size_t)(m0 + row) * (DEPTH * NHEAD) + kb);
#pragma unroll
    for (int t = 0; t < 4; ++t) {
      v2f b = mkv2(Wo[(kb + 0) * DEPTH + t * 16 + col],
                   Wo[(kb + 1) * DEPTH + t * 16 + col]);
      xs[t] = wmma4(a, b, xs[t]);
    }
  }
#pragma unroll
  for (int t = 0; t < 4; ++t)
#pragma unroll
    for (int r = 0; r < 8; ++r)
      xs[t][r] += seq[(m0 + r + 8 * hi) * DEPTH + t * 16 + col];

  float mu[8], rs[8];
#pragma unroll
  for (int r = 0; r < 8; ++r)
    mu[r] = hsum16(xs[0][r] + xs[1][r] + xs[2][r] + xs[3][r]) * (1.f / DEPTH);
#pragma unroll
  for (int r = 0; r < 8; ++r) {
    float s2 = 0.f;
#pragma unroll
    for (int t = 0; t < 4; ++t) {
      const float d0 = xs[t][r] - mu[r];
      s2 += d0 * d0;
    }
    rs[r] = rsqrtf(hsum16(s2) * (1.f / DEPTH) + 1e-5f);
  }
#pragma unroll
  for (int t = 0; t < 4; ++t)
#pragma unroll
    for (int r = 0; r < 8; ++r)
      xs[t][r] = (xs[t][r] - mu[r]) * rs[r] * g_att[t * 16 + col] +
                 b_att[t * 16 + col];

#pragma unroll
  for (int t = 0; t < 4; ++t)
#pragma unroll
    for (int r = 0; r < 8; ++r)
      xlds[(r + 8 * hi) * 68 + t * 16 + col] = xs[t][r];
  __syncthreads();

#pragma unroll
  for (int u = 0; u < 8; ++u) {
    v8f c = vzero8();
#pragma unroll
    for (int kk = 0; kk < 16; ++kk) {
      const int kb = 4 * kk + 2 * hi;
      v2f a = mkv2(xlds[row * 68 + kb], xlds[row * 68 + kb + 1]);
      v2f b = mkv2(W1[(kb + 0) * FFDIM + u * 16 + col],
                   W1[(kb + 1) * FFDIM + u * 16 + col]);
      c = wmma4(a, b, c);
    }
#pragma unroll
    for (int r = 0; r < 8; ++r)
      ylds[(r + 8 * hi) * 132 + u * 16 + col] =
          fmaxf(c[r] + b1[u * 16 + col], 0.f);
  }
  __syncthreads();

  v8f ys[4] = {vzero8(), vzero8(), vzero8(), vzero8()};
#pragma unroll
  for (int kk = 0; kk < 32; ++kk) {
    const int kb = 4 * kk + 2 * hi;
    v2f a = mkv2(ylds[row * 132 + kb], ylds[row * 132 + kb + 1]);
#pragma unroll
    for (int t = 0; t < 4; ++t) {
      v2f b = mkv2(W2[(kb + 0) * DEPTH + t * 16 + col],
                   W2[(kb + 1) * DEPTH + t * 16 + col]);
      ys[t] = wmma4(a, b, ys[t]);
    }
  }
#pragma unroll
  for (int t = 0; t < 4; ++t)
#pragma unroll
    for (int r = 0; r < 8; ++r) ys[t][r] += b2[t * 16 + col] + xs[t][r];

#pragma unroll
  for (int r = 0; r < 8; ++r)
    mu[r] = hsum16(ys[0][r] + ys[1][r] + ys[2][r] + ys[3][r]) * (1.f / DEPTH);
#pragma unroll
  for (int r = 0; r < 8; ++r) {
    float s2 = 0.f;
#pragma unroll
    for (int t = 0; t < 4; ++t) {
      const float d0 = ys[t][r] - mu[r];
      s2 += d0 * d0;
    }
    rs[r] = rsqrtf(hsum16(s2) * (1.f / DEPTH) + 1e-5f);
  }
#pragma unroll
  for (int t = 0; t < 4; ++t)
#pragma unroll
    for (int r = 0; r < 8; ++r)
      out[(m0 + r + 8 * hi) * DEPTH + t * 16 + col] =
          (ys[t][r] - mu[r]) * rs[r] * g_ff[t * 16 + col] + b_ff[t * 16 + col];
}

// ---------------------------------------------------------------------------
extern "C" void kernel_launch(void* const* d_in, const int* in_sizes, int n_in,
                              void* d_out, int out_size, void* d_ws,
                              size_t ws_size, hipStream_t stream) {
  const float* seq = (const float*)d_in[0];
  const float* pos = (const float*)d_in[1];
  const float* Wq = (const float*)d_in[2];
  const float* Wk = (const float*)d_in[3];
  const float* Wv = (const float*)d_in[4];
  const float* Wo = (const float*)d_in[5];
  const float* Wp = (const float*)d_in[6];
  const float* bp = (const float*)d_in[7];
  const float* W1 = (const float*)d_in[8];
  const float* b1 = (const float*)d_in[9];
  const float* W2 = (const float*)d_in[10];
  const float* b2 = (const float*)d_in[11];
  const float* g_att = (const float*)d_in[12];
  const float* b_att = (const float*)d_in[13];
  const float* g_ff = (const float*)d_in[14];
  const float* b_ff = (const float*)d_in[15];

  // Workspace (floats): q,k,v [B,h,S,d] (1M each), p [B,h,S,S] (8M),
  // o [B,S,h*d] (1M).  Total 48 MB.
  float* ws = (float*)d_ws;
  float* q = ws;
  float* k = q + (1 << 20);
  float* v = k + (1 << 20);
  float* p = v + (1 << 20);
  float* o = p + (8 << 20);

  qkv_kernel<<<dim3(BS / 16, (DEPTH * NHEAD) / 16, 3), 32, 0, stream>>>(
      seq, Wq, Wk, Wv, q, k, v);
#if defined(USE_ASYNC_LDS)
  posbias_kernel<<<dim3((BATCH * SEQ * SEQ) / 16 / POS_TILES_PER_BLOCK), 32, 0,
                   stream>>>(pos, Wp, bp, p);
#else
  posbias_kernel<<<dim3((BATCH * SEQ * SEQ) / 16), 32, 0, stream>>>(pos, Wp, bp,
                                                                    p);
#endif
  attn_kernel<<<dim3(SEQ / 16, BATCH * NHEAD), 32, 0, stream>>>(q, k, v, p, o);
  epilogue_kernel<<<dim3(BS / 16), 32, 0, stream>>>(
      o, seq, Wo, W1, b1, W2, b2, g_att, b_att, g_ff, b_ff, (float*)d_out);
}